// AtomicBasis_46952582480236
// MI455X (gfx1250) — compile-verified
//
#include <hip/hip_runtime.h>

typedef __attribute__((ext_vector_type(2))) float v2f;
typedef __attribute__((ext_vector_type(8))) float v8f;

// Fixed workload geometry (reference constants)
constexpr int kE = 400000;   // N_EDGES
constexpr int kN = 50000;    // N_NODES
constexpr int kG = 8;        // edges per wave

// ---------------------------------------------------------------------------
// Compile-time coefficient table: every entry of the per-edge 16x16 operator
// T(u) is a polynomial  T[r][c] = A0 + sum_a C_a u_a + sum_{a<=b} B_ab u_a u_b.
// Pair-major layout: v[lane][chunk t][slot][q] where (q=0,q=1) are the two
// entries of WMMA A-register pair t, i.e. T[lane&15][4t + q + 2*(lane>>4)].
// slots: [A0, C0, C1, C2, B00, B01, B02, B11, B12, B22]  (10 slots, 80B/chunk)
// ---------------------------------------------------------------------------
struct alignas(16) CoefTable { float v[32][4][10][2]; };

constexpr int bslot(int a, int b) {
    int lo = a < b ? a : b;
    int hi = a < b ? b : a;
    return (lo == 0) ? hi : (lo == 1 ? 2 + hi : 5);   // 00,01,02,11,12,22
}

constexpr CoefTable make_table() {
    CoefTable T{};
    for (int lane = 0; lane < 32; ++lane) {
        int r = lane & 15;
        int half = lane >> 4;
        for (int t = 0; t < 4; ++t) {
            for (int q = 0; q < 2; ++q) {
                int c = 4 * t + q + 2 * half;
                if (r >= 13 || c >= 13) continue;
                // V[s] += x  ==>  T.v[lane][t][s][q] += x
#define VADD(s, x) T.v[lane][t][s][q] += (x)
                int rt = (r == 0) ? 0 : (r <= 3 ? 1 : 2);
                int m = (rt == 1) ? (r - 1) : (rt == 2 ? (r - 4) / 3 : 0);
                int n = (rt == 2) ? (r - 4) % 3 : 0;
                int ct = (c == 0) ? 0 : (c <= 3 ? 1 : 2);
                int i = (ct == 1) ? (c - 1) : (ct == 2 ? (c - 4) / 3 : 0);
                int j = (ct == 2) ? (c - 4) % 3 : 0;

                if (rt == 0) {
                    if (ct == 0) { VADD(0,1.0f); VADD(4,1.0f); VADD(7,1.0f); VADD(9,1.0f); }
                    else if (ct == 1) { VADD(1 + i, 1.0f); }
                    else {
                        if (i == j) { VADD(0,1.0f); VADD(4,1.0f); VADD(7,1.0f); VADD(9,1.0f); }
                        VADD(4 + bslot(i, j), 2.0f);
                    }
                } else if (rt == 1) {
                    if (ct == 0) { VADD(1 + m, 1.0f); }
                    else if (ct == 1) {
                        if (i == m) { VADD(0,1.0f); VADD(4,1.0f); VADD(7,1.0f); VADD(9,1.0f); }
                        VADD(4 + bslot(m, i), 2.0f);
                    } else {
                        if (i == m) VADD(1 + j, 1.0f);
                        if (j == m) VADD(1 + i, 1.0f);
                        if (i == j) VADD(1 + m, 1.0f);
                    }
                } else {
                    if (ct == 0) { VADD(4 + bslot(m, n), 1.0f); }
                    else if (ct == 1) { if (i == m) VADD(1 + n, 1.0f); }
                    else {
                        if (i == m && j == n) { VADD(0,1.0f); VADD(4,1.0f); VADD(7,1.0f); VADD(9,1.0f); }
                        if (i == m) VADD(4 + bslot(j, n), 2.0f);
                        if (j == m) VADD(4 + bslot(i, n), 2.0f);
                        if (i == j) VADD(4 + bslot(m, n), 1.0f);
                    }
                }
#undef VADD
            }
        }
    }
    return T;
}

__constant__ CoefTable g_tab = make_table();

// Evaluate one WMMA A-register pair: 9 packed FMAs over the monomials.
__device__ __forceinline__ v2f chunk_eval(float4 q0, float4 q1, float4 q2,
                                          float4 q3, float4 q4,
                                          float u0, float u1, float u2,
                                          float p00, float p01, float p02,
                                          float p11, float p12, float p22) {
    v2f acc = {q0.x, q0.y};
    acc += (v2f){q0.z, q0.w} * u0;
    acc += (v2f){q1.x, q1.y} * u1;
    acc += (v2f){q1.z, q1.w} * u2;
    acc += (v2f){q2.x, q2.y} * p00;
    acc += (v2f){q2.z, q2.w} * p01;
    acc += (v2f){q3.x, q3.y} * p02;
    acc += (v2f){q3.z, q3.w} * p11;
    acc += (v2f){q4.x, q4.y} * p12;
    acc += (v2f){q4.z, q4.w} * p22;
    return acc;
}

// ---------------------------------------------------------------------------
// Kernel 0: zero the (accumulated) output
// ---------------------------------------------------------------------------
__global__ void zero_kernel(float4* __restrict__ p, int nelem4) {
    int i = blockIdx.x * blockDim.x + threadIdx.x;
    if (i < nelem4) p[i] = make_float4(0.f, 0.f, 0.f, 0.f);
}

// ---------------------------------------------------------------------------
// Kernel 1: per-edge geometry (unit vector + distance) into workspace.
// ---------------------------------------------------------------------------
__global__ void edge_geom_kernel(const float* __restrict__ pos, const int* __restrict__ ei,
                                 float4* __restrict__ geo) {
    int e = blockIdx.x * blockDim.x + threadIdx.x;
    if (e >= kE) return;
    int s = ei[e];
    int d = ei[kE + e];
    float rx = pos[3 * s + 0] - pos[3 * d + 0];
    float ry = pos[3 * s + 1] - pos[3 * d + 1];
    float rz = pos[3 * s + 2] - pos[3 * d + 2];
    float rr = sqrtf(rx * rx + ry * ry + rz * rz);
    float inv = 1.0f / rr;
    geo[e] = make_float4(rx * inv, ry * inv, rz * inv, rr);
}

// ---------------------------------------------------------------------------
// Kernel 2: one wave handles kG consecutive edges; coefficient table is
// loaded to registers ONCE per wave.  Per edge: D = T(u) x H via 4 chained
// v_wmma_f32_16x16x4_f32, scale columns by R[e,k], atomic scatter to src.
// ---------------------------------------------------------------------------
__global__ __launch_bounds__(256) void edge_wmma_kernel(
    const float4* __restrict__ geo, const int* __restrict__ ei,
    const float* __restrict__ h0, const float* __restrict__ h1,
    const float* __restrict__ h2, float* __restrict__ out)
{
    int wave = blockIdx.x * 8 + (threadIdx.x >> 5);
    int base = wave * kG;

    int lane = threadIdx.x & 31;
    int half = lane >> 4;                         // 0 or 1
    int k    = lane & 15;                         // A row AND B/D channel column

    // Hoist the whole per-lane coefficient set into registers (20 x b128).
    const float4* tp = (const float4*)&g_tab.v[lane][0][0][0];
    float4 c00 = tp[0],  c01 = tp[1],  c02 = tp[2],  c03 = tp[3],  c04 = tp[4];
    float4 c10 = tp[5],  c11 = tp[6],  c12 = tp[7],  c13 = tp[8],  c14 = tp[9];
    float4 c20 = tp[10], c21 = tp[11], c22 = tp[12], c23 = tp[13], c24 = tp[14];
    float4 c30 = tp[15], c31 = tp[16], c32 = tp[17], c33 = tp[18], c34 = tp[19];

    for (int it = 0; it < kG; ++it) {
        int eu = __builtin_amdgcn_readfirstlane(base + it);   // uniform edge id

        int src = ei[eu];                         // uniform -> s_load
        int dst = ei[kE + eu];
        float4 g = geo[eu];                       // uniform -> s_load_b128
        float u0 = g.x, u1 = g.y, u2 = g.z;
        float p00 = u0 * u0, p01 = u0 * u1, p02 = u0 * u2;
        float p11 = u1 * u1, p12 = u1 * u2, p22 = u2 * u2;

        v2f a0 = chunk_eval(c00, c01, c02, c03, c04, u0, u1, u2, p00, p01, p02, p11, p12, p22);
        v2f a1 = chunk_eval(c10, c11, c12, c13, c14, u0, u1, u2, p00, p01, p02, p11, p12, p22);
        v2f a2 = chunk_eval(c20, c21, c22, c23, c24, u0, u1, u2, p00, p01, p02, p11, p12, p22);
        v2f a3 = chunk_eval(c30, c31, c32, c33, c34, u0, u1, u2, p00, p01, p02, p11, p12, p22);

        // B fragments: H[j = 4t + q + 2*half, k], 32-bit offsets off uniform bases
        int hb1 = dst * 48 + k * 3;        // h1[dst,k,0]
        int hb2 = dst * 144 + k * 9;       // h2[dst,k,0,0]
        int ho  = 2 * half;
        float vh0  = h0[dst * 16 + k];     // j=0  (half 0)
        float vh1a = h1[hb1 + 1];          // j=2  (half 1)
        v2f b0 = { half ? vh1a : vh0,      // j = 0 | 2
                   h1[hb1 + ho] };         // j = 1 | 3
        v2f b1 = { h2[hb2 + ho],           // j = 4 | 6
                   h2[hb2 + 1 + ho] };     // j = 5 | 7
        v2f b2 = { h2[hb2 + 4 + ho],       // j = 8 | 10
                   h2[hb2 + 5 + ho] };     // j = 9 | 11
        float vh2t = h2[hb2 + 8];          // j=12 (half 0 only)
        v2f b3 = { half ? 0.0f : vh2t, 0.0f };

        v8f d = {0.f, 0.f, 0.f, 0.f, 0.f, 0.f, 0.f, 0.f};
        d = __builtin_amdgcn_wmma_f32_16x16x4_f32(false, a0, false, b0, (short)0, d, false, false);
        d = __builtin_amdgcn_wmma_f32_16x16x4_f32(false, a1, false, b1, (short)0, d, false, false);
        d = __builtin_amdgcn_wmma_f32_16x16x4_f32(false, a2, false, b2, (short)0, d, false, false);
        d = __builtin_amdgcn_wmma_f32_16x16x4_f32(false, a3, false, b3, (short)0, d, false, false);

        // Radial weight with the faithful [K,E]->[E,K] reshape: iflat = 16e+k.
        int base16 = eu * 16;
        int qn  = base16 / kE;                     // SALU magic multiply
        int rem = base16 - qn * kE;
        int tt  = rem + k;
        int carry = (tt >= kE) ? 1 : 0;
        int nidx  = qn + carry + 1;                // n value (1-based)
        int ecol  = tt - (carry ? kE : 0);
        float rr = ((const float*)geo)[4 * ecol + 3];
        float sv = __sinf((float)nidx * 0.31415926535897931f * rr);
        float Rv = 0.4472135954999579f * sv * __builtin_amdgcn_rcpf(rr);

        // Scatter with 32-bit element offsets off the single uniform out base.
        int idx0 = src * 16 + k;                   // res0 slot
        int idx1 = 16 * kN + idx0 * 3;             // res1[...,0]
        int idx2 = 64 * kN + idx0 * 9;             // res2[...,0]

        atomicAdd(out + (half ? idx2 + 4 : idx0),     d[0] * Rv);  // row 0 | 8
        atomicAdd(out + (half ? idx2 + 5 : idx1),     d[1] * Rv);  // row 1 | 9
        atomicAdd(out + (half ? idx2 + 6 : idx1 + 1), d[2] * Rv);  // row 2 | 10
        atomicAdd(out + (half ? idx2 + 7 : idx1 + 2), d[3] * Rv);  // row 3 | 11
        atomicAdd(out + idx2 + 8 * half,              d[4] * Rv);  // row 4 | 12
        if (!half) {                                                // rows 5,6,7
            atomicAdd(out + idx2 + 1, d[5] * Rv);
            atomicAdd(out + idx2 + 2, d[6] * Rv);
            atomicAdd(out + idx2 + 3, d[7] * Rv);
        }
    }
}

// ---------------------------------------------------------------------------
extern "C" void kernel_launch(void* const* d_in, const int* in_sizes, int n_in,
                              void* d_out, int out_size, void* d_ws, size_t ws_size,
                              hipStream_t stream) {
    (void)in_sizes; (void)n_in; (void)ws_size;
    const float* h0  = (const float*)d_in[0];
    const float* h1  = (const float*)d_in[1];
    const float* h2  = (const float*)d_in[2];
    const float* pos = (const float*)d_in[3];
    // d_in[4] = channel_weights: dead in the reference forward (replicated)
    const int*   ei  = (const int*)d_in[5];

    float*  out = (float*)d_out;
    float4* geo = (float4*)d_ws;   // 16 B/edge = 6.4 MB

    int nelem4 = out_size / 4;     // 208*kN divisible by 4
    zero_kernel<<<(nelem4 + 255) / 256, 256, 0, stream>>>((float4*)out, nelem4);
    edge_geom_kernel<<<(kE + 255) / 256, 256, 0, stream>>>(pos, ei, geo);
    // kG edges per wave, 8 waves per block: kE/(kG*8) = 6250 blocks exactly
    edge_wmma_kernel<<<kE / (kG * 8), 256, 0, stream>>>(geo, ei, h0, h1, h2, out);
}